// PrecomputedMetaNet_81836306858222
// MI455X (gfx1250) — compile-verified
//
#include <hip/hip_runtime.h>

// ---------------------------------------------------------------------------
// Problem constants (from the reference):
//   B = 2048 rows, D = 1024, H = 256, T = 8, NUM_BLOCKS = 12
// ---------------------------------------------------------------------------
#define BDIM   2048
#define DDIM   1024
#define HDIM   256
#define TDIM   8
#define NBLK   12

typedef __attribute__((ext_vector_type(2))) float v2f;
typedef __attribute__((ext_vector_type(8))) float v8f;

// ----------------------------------------------------------------------------
// WMMA GEMM tile configuration.
//   Block tile: 128 (M) x 128 (N), K-chunk 16, double-buffered LDS.
//   256 threads = 8 waves, arranged 4 (M) x 2 (N); each wave computes a
//   32x64 patch = 2x4 tiles of V_WMMA_F32_16X16X4_F32 (64 accum VGPRs).
//   Both LDS tiles are stored K-minor with stride 17 so every fragment is a
//   contiguous dword pair -> single ds_load_2addr_b32, no repack movs.
// ----------------------------------------------------------------------------
#define BM   128
#define BN   128
#define BK   16
#define LDK  (BK + 1)   // padded K stride (floats) for both LDS tiles

// MODE: 0 = plain store (out = A@B)
//       1 = chain step  (out = A + coeff[row] * (A@B)); requires N == K
//       2 = bias + ReLU (out = relu(A@B + bias[col]))
template <int MODE>
__global__ __launch_bounds__(256)
void gemm_f32_wmma(const float* __restrict__ A,    // [M, K] row-major
                   const float* __restrict__ Bsrc, // transB=0: [K, N]; transB=1: [N, K]
                   float* __restrict__ out,        // [M, N]
                   int M, int N, int K, int transB,
                   const float* __restrict__ coeff, int cstride, int ccol,
                   const float* __restrict__ bias)
{
    __shared__ float ldsA[2][BM * LDK];   // [row][k]
    __shared__ float ldsB[2][BN * LDK];   // [n][k]  (transposed tile)

    const int tid  = threadIdx.x;
    const int lane = tid & 31;
    const int wave = tid >> 5;
    const int lm   = lane & 15;        // M (A frag) / N (B frag) within tile
    const int half = lane >> 4;        // selects K pair {0,1} vs {2,3}
    const int wm   = (wave >> 1) * 32; // wave M offset in block tile
    const int wn   = (wave & 1) * 64;  // wave N offset in block tile

    const int m0 = blockIdx.y * BM;
    const int n0 = blockIdx.x * BN;

    // Staging decomposition (per thread, 2 float4 for A and 2 for B per chunk)
    const int aRow = tid >> 2;               // 4 float4 per 16-wide A row
    const int aCol = (tid & 3) << 2;
    const int bnRowK = tid >> 5;             // NN: 32 float4 per 128-wide B row
    const int bnColN = (tid & 31) << 2;
    const int btRowN = tid >> 2;             // NT: 4 float4 per 16-wide P row
    const int btColK = (tid & 3) << 2;

    const v8f vz = {0.f,0.f,0.f,0.f,0.f,0.f,0.f,0.f};
    v8f acc[2][4];
    #pragma unroll
    for (int mt = 0; mt < 2; ++mt)
        #pragma unroll
        for (int nt = 0; nt < 4; ++nt) acc[mt][nt] = vz;

    const int nchunks = K / BK;
    float4 ra[2], rb[2];

    // ---- prefetch chunk 0 ----
    {
        const int k0 = 0;
        #pragma unroll
        for (int i = 0; i < 2; ++i)
            ra[i] = *(const float4*)(A + (size_t)(m0 + aRow + i * 64) * K + k0 + aCol);
        if (!transB) {
            #pragma unroll
            for (int i = 0; i < 2; ++i)
                rb[i] = *(const float4*)(Bsrc + (size_t)(k0 + bnRowK + i * 8) * N + n0 + bnColN);
        } else {
            #pragma unroll
            for (int i = 0; i < 2; ++i)
                rb[i] = *(const float4*)(Bsrc + (size_t)(n0 + btRowN + i * 64) * K + k0 + btColK);
        }
    }
    // ---- store chunk 0 to LDS buf 0 ----
    #pragma unroll
    for (int i = 0; i < 2; ++i) {
        float* d = &ldsA[0][(aRow + i * 64) * LDK + aCol];
        d[0] = ra[i].x; d[1] = ra[i].y; d[2] = ra[i].z; d[3] = ra[i].w;
    }
    if (!transB) {
        #pragma unroll
        for (int i = 0; i < 2; ++i) {
            float* d = &ldsB[0][bnColN * LDK + bnRowK + i * 8];
            d[0 * LDK] = rb[i].x; d[1 * LDK] = rb[i].y;
            d[2 * LDK] = rb[i].z; d[3 * LDK] = rb[i].w;
        }
    } else {
        #pragma unroll
        for (int i = 0; i < 2; ++i) {
            float* d = &ldsB[0][(btRowN + i * 64) * LDK + btColK];
            d[0] = rb[i].x; d[1] = rb[i].y; d[2] = rb[i].z; d[3] = rb[i].w;
        }
    }
    __syncthreads();

    for (int c = 0; c < nchunks; ++c) {
        const int buf = c & 1;

        // ---- issue global prefetch of chunk c+1 (latency hidden by WMMAs) ----
        const bool more = (c + 1) < nchunks;
        if (more) {
            const int k0 = (c + 1) * BK;
            #pragma unroll
            for (int i = 0; i < 2; ++i)
                ra[i] = *(const float4*)(A + (size_t)(m0 + aRow + i * 64) * K + k0 + aCol);
            if (!transB) {
                #pragma unroll
                for (int i = 0; i < 2; ++i)
                    rb[i] = *(const float4*)(Bsrc + (size_t)(k0 + bnRowK + i * 8) * N + n0 + bnColN);
            } else {
                #pragma unroll
                for (int i = 0; i < 2; ++i)
                    rb[i] = *(const float4*)(Bsrc + (size_t)(n0 + btRowN + i * 64) * K + k0 + btColK);
            }
        }

        // ---- compute: 4 K-steps, fragments double-buffered in registers ----
        v2f af[2][2], bf[2][4];
        {   // load fragment group 0
            const int kc = 2 * half;
            #pragma unroll
            for (int mt = 0; mt < 2; ++mt) {
                const float* p = &ldsA[buf][(wm + mt * 16 + lm) * LDK + kc];
                af[0][mt].x = p[0]; af[0][mt].y = p[1];
            }
            #pragma unroll
            for (int nt = 0; nt < 4; ++nt) {
                const float* p = &ldsB[buf][(wn + nt * 16 + lm) * LDK + kc];
                bf[0][nt].x = p[0]; bf[0][nt].y = p[1];
            }
        }
        #pragma unroll
        for (int kk = 0; kk < 4; ++kk) {
            const int cur = kk & 1, nxt = cur ^ 1;
            if (kk < 3) {       // prefetch next fragment group
                const int kc = (kk + 1) * 4 + 2 * half;
                #pragma unroll
                for (int mt = 0; mt < 2; ++mt) {
                    const float* p = &ldsA[buf][(wm + mt * 16 + lm) * LDK + kc];
                    af[nxt][mt].x = p[0]; af[nxt][mt].y = p[1];
                }
                #pragma unroll
                for (int nt = 0; nt < 4; ++nt) {
                    const float* p = &ldsB[buf][(wn + nt * 16 + lm) * LDK + kc];
                    bf[nxt][nt].x = p[0]; bf[nxt][nt].y = p[1];
                }
            }
            #pragma unroll
            for (int mt = 0; mt < 2; ++mt)
                #pragma unroll
                for (int nt = 0; nt < 4; ++nt)
                    acc[mt][nt] = __builtin_amdgcn_wmma_f32_16x16x4_f32(
                        false, af[cur][mt], false, bf[cur][nt],
                        (short)0, acc[mt][nt], false, false);
        }

        // ---- stage chunk c+1 into the other LDS buffer ----
        if (more) {
            const int nb = buf ^ 1;
            #pragma unroll
            for (int i = 0; i < 2; ++i) {
                float* d = &ldsA[nb][(aRow + i * 64) * LDK + aCol];
                d[0] = ra[i].x; d[1] = ra[i].y; d[2] = ra[i].z; d[3] = ra[i].w;
            }
            if (!transB) {
                #pragma unroll
                for (int i = 0; i < 2; ++i) {
                    float* d = &ldsB[nb][bnColN * LDK + bnRowK + i * 8];
                    d[0 * LDK] = rb[i].x; d[1 * LDK] = rb[i].y;
                    d[2 * LDK] = rb[i].z; d[3 * LDK] = rb[i].w;
                }
            } else {
                #pragma unroll
                for (int i = 0; i < 2; ++i) {
                    float* d = &ldsB[nb][(btRowN + i * 64) * LDK + btColK];
                    d[0] = rb[i].x; d[1] = rb[i].y; d[2] = rb[i].z; d[3] = rb[i].w;
                }
            }
        }
        __syncthreads();
    }

    // ---- epilogue: C/D layout -> VGPR v holds row (v + 8*half), col = lm ----
    #pragma unroll
    for (int mt = 0; mt < 2; ++mt) {
        #pragma unroll
        for (int v = 0; v < 8; ++v) {
            const int grow = m0 + wm + mt * 16 + v + half * 8;
            float cf = 0.f;
            if (MODE == 1) cf = coeff[(size_t)grow * cstride + ccol];
            #pragma unroll
            for (int nt = 0; nt < 4; ++nt) {
                const int gcol = n0 + wn + nt * 16 + lm;
                float val = acc[mt][nt][v];
                if (MODE == 0) {
                    out[(size_t)grow * N + gcol] = val;
                } else if (MODE == 1) {
                    const float x = A[(size_t)grow * K + gcol];   // N == K here
                    out[(size_t)grow * N + gcol] = fmaf(cf, val, x);
                } else {
                    val += bias[gcol];
                    out[(size_t)grow * N + gcol] = fmaxf(val, 0.0f);
                }
            }
        }
    }
}

// ----------------------------------------------------------------------------
// Fold the 12-block structure of (w2, b2): mean over blocks commutes with the
// linear layer.  w2m[t][h] = mean_j w2[t*12+j][h], b2m[t] = mean_j b2[t*12+j].
// ----------------------------------------------------------------------------
__global__ void reduce_w2_kernel(const float* __restrict__ w2,
                                 const float* __restrict__ b2,
                                 float* __restrict__ w2m,
                                 float* __restrict__ b2m)
{
    const int hh = threadIdx.x;             // 256 threads
    const float inv = 1.0f / (float)NBLK;
    #pragma unroll
    for (int t = 0; t < TDIM; ++t) {
        float s = 0.f;
        #pragma unroll
        for (int j = 0; j < NBLK; ++j)
            s += w2[(size_t)(t * NBLK + j) * HDIM + hh];
        w2m[t * HDIM + hh] = s * inv;
    }
    if (hh < TDIM) {
        float s = 0.f;
        #pragma unroll
        for (int j = 0; j < NBLK; ++j) s += b2[hh * NBLK + j];
        b2m[hh] = s * inv;
    }
}

// ----------------------------------------------------------------------------
// coeffs[b][t] = h[b,:] . w2m[t,:] + b2m[t]   (N=8 is too skinny for WMMA)
// ----------------------------------------------------------------------------
__global__ __launch_bounds__(256)
void coeffs_kernel(const float* __restrict__ h,
                   const float* __restrict__ w2m,
                   const float* __restrict__ b2m,
                   float* __restrict__ coeffs)
{
    const int idx = blockIdx.x * blockDim.x + threadIdx.x;  // [0, B*T)
    const int b = idx >> 3;
    const int t = idx & 7;
    const float4* hp = (const float4*)(h + (size_t)b * HDIM);
    const float4* wp = (const float4*)(w2m + (size_t)t * HDIM);
    float s = 0.f;
    #pragma unroll 8
    for (int i = 0; i < HDIM / 4; ++i) {
        const float4 hv = hp[i];
        const float4 wv = wp[i];
        s = fmaf(hv.x, wv.x, s);
        s = fmaf(hv.y, wv.y, s);
        s = fmaf(hv.z, wv.z, s);
        s = fmaf(hv.w, wv.w, s);
    }
    coeffs[idx] = s + b2m[t];
}

// ----------------------------------------------------------------------------
extern "C" void kernel_launch(void* const* d_in, const int* in_sizes, int n_in,
                              void* d_out, int out_size, void* d_ws, size_t ws_size,
                              hipStream_t stream)
{
    const float* features  = (const float*)d_in[0];  // [2048, 1024]
    const float* w1        = (const float*)d_in[1];  // [256, 1024]
    const float* b1        = (const float*)d_in[2];  // [256]
    const float* w2        = (const float*)d_in[3];  // [96, 256]
    const float* b2        = (const float*)d_in[4];  // [96]
    const float* task_mats = (const float*)d_in[5];  // [8, 1024, 1024]
    const float* proj_w    = (const float*)d_in[6];  // [1024, 1024]
    float* out = (float*)d_out;                      // [2048, 1024]

    // Workspace layout (floats)
    float* ws     = (float*)d_ws;
    float* w2m    = ws;                       // 8*256      = 2048
    float* b2m    = w2m + TDIM * HDIM;        // 8 (pad 64)
    float* h      = b2m + 64;                 // 2048*256   = 524288
    float* coeffs = h + (size_t)BDIM * HDIM;  // 2048*8     = 16384
    float* x0     = coeffs + (size_t)BDIM * TDIM;        // 2048*1024
    float* x1     = x0 + (size_t)BDIM * DDIM;            // 2048*1024

    // 1) Fold the 12-block meta-net output layer.
    reduce_w2_kernel<<<1, 256, 0, stream>>>(w2, b2, w2m, b2m);

    // 2) h = relu(features @ w1^T + b1)   [2048 x 256], NT GEMM, bias+ReLU.
    gemm_f32_wmma<2><<<dim3(HDIM / BN, BDIM / BM), 256, 0, stream>>>(
        features, w1, h, BDIM, HDIM, DDIM, /*transB=*/1,
        nullptr, 0, 0, b1);

    // 3) coeffs = h @ w2m^T + b2m        [2048 x 8]
    coeffs_kernel<<<(BDIM * TDIM) / 256, 256, 0, stream>>>(h, w2m, b2m, coeffs);

    // 4) Sequential chain: x <- x + c_j * (x @ M_j), j = 0..7 (NN GEMMs).
    const float* xin = features;
    float* xout = x0;
    for (int j = 0; j < TDIM; ++j) {
        gemm_f32_wmma<1><<<dim3(DDIM / BN, BDIM / BM), 256, 0, stream>>>(
            xin, task_mats + (size_t)j * DDIM * DDIM, xout,
            BDIM, DDIM, DDIM, /*transB=*/0,
            coeffs, TDIM, j, nullptr);
        xin  = xout;
        xout = (xout == x0) ? x1 : x0;
    }

    // 5) out = x @ proj_w^T              [2048 x 1024], NT GEMM, plain store.
    gemm_f32_wmma<0><<<dim3(DDIM / BN, BDIM / BM), 256, 0, stream>>>(
        xin, proj_w, out, BDIM, DDIM, DDIM, /*transB=*/1,
        nullptr, 0, 0, nullptr);
}